// TopKSAE_8615704395899
// MI455X (gfx1250) — compile-verified
//
#include <hip/hip_runtime.h>

typedef float v2f __attribute__((ext_vector_type(2)));
typedef float v8f __attribute__((ext_vector_type(8)));

constexpr int D_IN  = 1024;
constexpr int D_SAE = 16384;
constexpr int NROWS = 4096;
constexpr int K_TOP = 32;

// GEMM tiling: block = 256 thr (8 waves in 2x4), tile 64(M) x 256(N), k-slab 16.
// Wave strip = 32(M) x 64(N) = 2x4 subtiles of 16x16.
constexpr int BM = 64;
constexpr int BN = 256;
constexpr int BK = 16;
constexpr int KPAIRS = BK / 2;           // 8 k-pairs staged per slab
constexpr int LDA2 = 80;                 // float2 stride per k-pair row of A (2*80 mod 64 == 32)
constexpr int LDB2 = 272;                // float2 stride per k-pair row of B (2*272 mod 64 == 32)

// ---- CDNA5 async memory->LDS helpers (ASYNCcnt-tracked) --------------------
__device__ __forceinline__ void async_copy_b32(unsigned lds_off, const void* g) {
    asm volatile("global_load_async_to_lds_b32 %0, %1, off"
                 :: "v"(lds_off), "v"(g) : "memory");
}
__device__ __forceinline__ void async_copy_b64(unsigned lds_off, const void* g) {
    asm volatile("global_load_async_to_lds_b64 %0, %1, off"
                 :: "v"(lds_off), "v"(g) : "memory");
}
__device__ __forceinline__ void wait_async0() {
    asm volatile("s_wait_asynccnt 0x0" ::: "memory");
}
// Flat shared-aperture address keeps the wave-relative LDS offset in addr[31:0].
__device__ __forceinline__ unsigned lds_off(const void* p) {
    return (unsigned)(unsigned long long)p;
}

// ---------------------------------------------------------------------------
// Kernel 0: benc_eff[c] = b_enc[c] - sum_k b_dec[k] * W_enc[k, c]
// Lets the GEMM A-path be a pure copy (async-friendly):
//   (x - b_dec) @ W_enc + b_enc  ==  x @ W_enc + benc_eff
// ---------------------------------------------------------------------------
__global__ __launch_bounds__(256)
void sae_bias_fold(const float* __restrict__ W_enc,   // [D_IN, D_SAE]
                   const float* __restrict__ b_enc,   // [D_SAE]
                   const float* __restrict__ b_dec,   // [D_IN]
                   float* __restrict__ benc_eff)      // [D_SAE]
{
    const int c = blockIdx.x * 256 + threadIdx.x;
    float s = 0.f;
    for (int k = 0; k < D_IN; ++k)
        s += b_dec[k] * W_enc[(size_t)k * D_SAE + c];   // coalesced across threads
    benc_eff[c] = b_enc[c] - s;
}

// ---------------------------------------------------------------------------
// Kernel 1: pre = x @ W_enc + benc_eff   via V_WMMA_F32_16X16X4_F32
// Double-buffered LDS slabs filled with GLOBAL_LOAD_ASYNC_TO_LDS; compute on
// one buffer while the TD/async engine fills the other. Pair-major LDS layout
// so every WMMA fragment is one aligned ds_load_b64.
// ---------------------------------------------------------------------------
__global__ __launch_bounds__(256)
void sae_encode_gemm(const float* __restrict__ x,        // [NROWS, D_IN]
                     const float* __restrict__ W_enc,    // [D_IN, D_SAE]
                     const float* __restrict__ benc_eff, // [D_SAE]
                     float* __restrict__ pre)            // [NROWS, D_SAE]
{
    __shared__ v2f Asp[2][KPAIRS * LDA2];   // [buf][kpair][m]  m = 0..63
    __shared__ v2f Bsp[2][KPAIRS * LDB2];   // [buf][kpair][c]  c = 0..255

    const int tid     = threadIdx.x;
    const int rowBase = blockIdx.y * BM;
    const int colBase = blockIdx.x * BN;
    const int wid     = tid >> 5;
    const int lane    = tid & 31;
    const int half    = lane >> 4;         // selects K-pair p (0) vs p+1 (1)
    const int l16     = lane & 15;         // M (A) / N (B,C,D) inside a 16-wide subtile
    const int wm      = (wid & 1) * 32;    // wave's M offset in block tile
    const int wn      = (wid >> 1) * 64;   // wave's N offset in block tile

    const v8f zero = {0.f, 0.f, 0.f, 0.f, 0.f, 0.f, 0.f, 0.f};
    v8f acc[2][4];
    #pragma unroll
    for (int mt = 0; mt < 2; ++mt)
        #pragma unroll
        for (int nt = 0; nt < 4; ++nt)
            acc[mt][nt] = zero;

    // Issue async staging of slab kb into LDS buffer `buf`.
    auto stage = [&](int kb, int buf) {
        // A: 64 rows x 8 k-pairs; contiguous K-pair in x -> async b64
        #pragma unroll
        for (int i = 0; i < 2; ++i) {
            int idx = tid + i * 256;       // 0..511
            int kp  = idx & 7;
            int m   = idx >> 3;
            const float* g = x + (size_t)(rowBase + m) * D_IN + kb + 2 * kp;
            async_copy_b64(lds_off(&Asp[buf][kp * LDA2 + m]), g);
        }
        // B: 8 k-pairs x 256 cols; pair-transpose -> two async b32 per float2
        #pragma unroll
        for (int i = 0; i < 8; ++i) {
            int idx = tid + i * 256;       // 0..2047
            int c   = idx & 255;
            int kp  = idx >> 8;
            const float* g = W_enc + (size_t)(kb + 2 * kp) * D_SAE + colBase + c;
            unsigned l = lds_off(&Bsp[buf][kp * LDB2 + c]);
            async_copy_b32(l,     g);          // k = kb+2kp   -> .x
            async_copy_b32(l + 4, g + D_SAE);  // k = kb+2kp+1 -> .y
        }
    };

    stage(0, 0);

    for (int kb = 0; kb < D_IN; kb += BK) {
        const int buf = (kb / BK) & 1;
        // Current buffer's async fills must land; all waves' LDS reads of the
        // buffer we are about to overwrite finished before this barrier.
        wait_async0();
        __syncthreads();
        if (kb + BK < D_IN)
            stage(kb + BK, buf ^ 1);   // overlap next slab's fill with compute

        #pragma unroll
        for (int kk = 0; kk < BK; kk += 4) {
            const int p = (kk >> 1) + half;  // lane-half picks K-pair p / p+1
            v2f a[2], b[4];
            #pragma unroll
            for (int mt = 0; mt < 2; ++mt)
                a[mt] = Asp[buf][p * LDA2 + wm + mt * 16 + l16];
            #pragma unroll
            for (int nt = 0; nt < 4; ++nt)
                b[nt] = Bsp[buf][p * LDB2 + wn + nt * 16 + l16];
            #pragma unroll
            for (int mt = 0; mt < 2; ++mt)
                #pragma unroll
                for (int nt = 0; nt < 4; ++nt)
                    acc[mt][nt] = __builtin_amdgcn_wmma_f32_16x16x4_f32(
                        /*neg_a=*/false, a[mt], /*neg_b=*/false, b[nt],
                        /*c_mod=*/(short)0, acc[mt][nt],
                        /*reuse_a=*/false, /*reuse_b=*/false);
        }
    }

    // Store C (+ benc_eff): VGPR r -> M = r + 8*half, N = l16
    #pragma unroll
    for (int mt = 0; mt < 2; ++mt) {
        #pragma unroll
        for (int nt = 0; nt < 4; ++nt) {
            int col  = colBase + wn + nt * 16 + l16;
            float be = benc_eff[col];
            #pragma unroll
            for (int r = 0; r < 8; ++r) {
                int row = rowBase + wm + mt * 16 + r + 8 * half;
                pre[(size_t)row * D_SAE + col] = acc[mt][nt][r] + be;
            }
        }
    }
}

// ---------------------------------------------------------------------------
// Kernel 2: per-row top-32 select (iterative argmax, tie -> lowest index),
// sparsify z in-place, then decode x_rec = z @ W_dec + b_dec.
// One block (256 thr) per row; row cached in registers (64 elems/thread).
// ---------------------------------------------------------------------------
__global__ __launch_bounds__(256)
void sae_topk_decode(float* __restrict__ z,            // [NROWS, D_SAE] in: pre, out: z_sparse
                     const float* __restrict__ W_dec,  // [D_SAE, D_IN]
                     const float* __restrict__ b_dec,  // [D_IN]
                     float* __restrict__ xrec)         // [NROWS, D_IN]
{
    constexpr int SLOTS = D_SAE / 256;  // 64 elements per thread
    __shared__ float red_val[256];
    __shared__ int   red_idx[256];
    __shared__ float topv[K_TOP];
    __shared__ int   topi[K_TOP];

    const int row = blockIdx.x;
    const int tid = threadIdx.x;
    const float NEG = -__builtin_inff();

    float local[SLOTS];
    const size_t zbase = (size_t)row * D_SAE;
    #pragma unroll
    for (int i = 0; i < SLOTS; ++i)
        local[i] = z[zbase + tid + (i << 8)];

    for (int it = 0; it < K_TOP; ++it) {
        float lv = NEG;
        int   li = 0x7fffffff;
        #pragma unroll
        for (int i = 0; i < SLOTS; ++i) {
            float v = local[i];
            int   j = tid + (i << 8);
            if (v > lv || (v == lv && j < li)) { lv = v; li = j; }
        }
        red_val[tid] = lv;
        red_idx[tid] = li;
        __syncthreads();
        for (int s = 128; s > 0; s >>= 1) {
            if (tid < s) {
                float ov = red_val[tid + s]; int oi = red_idx[tid + s];
                float mv = red_val[tid];     int mi = red_idx[tid];
                if (ov > mv || (ov == mv && oi < mi)) {
                    red_val[tid] = ov; red_idx[tid] = oi;
                }
            }
            __syncthreads();
        }
        if (tid == 0) { topv[it] = red_val[0]; topi[it] = red_idx[0]; }
        __syncthreads();
        int wj = topi[it];
        if ((wj & 255) == tid) local[wj >> 8] = NEG;
    }
    __syncthreads();

    // Single-pass sparsify: 0 everywhere except selected -> relu(value).
    #pragma unroll 4
    for (int i = 0; i < SLOTS; ++i) {
        int j = tid + (i << 8);
        float out = 0.f;
        if (local[i] == NEG) {
            #pragma unroll
            for (int t = 0; t < K_TOP; ++t)
                if (topi[t] == j) out = fmaxf(topv[t], 0.f);
        }
        z[zbase + j] = out;
    }

    // Decode: x_rec[row] = sum_t relu(topv[t]) * W_dec[topi[t], :] + b_dec
    const float4* bd4 = (const float4*)b_dec;
    float4 acc4 = bd4[tid];
    for (int t = 0; t < K_TOP; ++t) {
        float v = fmaxf(topv[t], 0.f);
        const float4* w4 = (const float4*)(W_dec + (size_t)topi[t] * D_IN);
        float4 w = w4[tid];
        acc4.x += v * w.x;
        acc4.y += v * w.y;
        acc4.z += v * w.z;
        acc4.w += v * w.w;
    }
    ((float4*)(xrec + (size_t)row * D_IN))[tid] = acc4;
}

// ---------------------------------------------------------------------------
extern "C" void kernel_launch(void* const* d_in, const int* in_sizes, int n_in,
                              void* d_out, int out_size, void* d_ws, size_t ws_size,
                              hipStream_t stream) {
    const float* x     = (const float*)d_in[0];
    const float* W_enc = (const float*)d_in[1];
    const float* b_enc = (const float*)d_in[2];
    const float* W_dec = (const float*)d_in[3];
    const float* b_dec = (const float*)d_in[4];
    // d_out = [x_reconstruct (NROWS*D_IN), z_sparse (NROWS*D_SAE)]
    float* xrec     = (float*)d_out;
    float* z        = (float*)d_out + (size_t)NROWS * D_IN;
    float* benc_eff = (float*)d_ws;   // 64 KB scratch

    sae_bias_fold<<<D_SAE / 256, 256, 0, stream>>>(W_enc, b_enc, b_dec, benc_eff);
    dim3 grid(D_SAE / BN, NROWS / BM);
    sae_encode_gemm<<<grid, 256, 0, stream>>>(x, W_enc, benc_eff, z);
    sae_topk_decode<<<NROWS, 256, 0, stream>>>(z, W_dec, b_dec, xrec);
}